// GNN_36816459661380
// MI455X (gfx1250) — compile-verified
//
#include <hip/hip_runtime.h>
#include <hip/hip_bf16.h>
#include <stdint.h>

typedef __attribute__((ext_vector_type(2))) float v2f;
typedef __attribute__((ext_vector_type(8))) float v8f;

#define D 256            // feature dim (D_IN == HID == OUT == 256)
#define TILE_ROWS 128    // rows per block (8 waves x 16)
#define TILE_COLS 128    // cols per block (8 wmma tiles per wave)
#define KCHUNK    64     // K rows of weights staged in LDS per pass (2x32KB = 64KB)

// ---------------------------------------------------------------------------
// relaxed agent-scope f32 atomic add -> global_atomic_add_f32 (no return)
// ---------------------------------------------------------------------------
__device__ __forceinline__ void atomAddF(float* p, float v) {
    __hip_atomic_fetch_add(p, v, __ATOMIC_RELAXED, __HIP_MEMORY_SCOPE_AGENT);
}

// ---------------------------------------------------------------------------
// async copy of 16B from global to LDS (per-lane), tracked by ASYNCcnt
// ---------------------------------------------------------------------------
__device__ __forceinline__ void asyncCopy16(unsigned lds_byte_addr, const float* gsrc) {
    asm volatile("global_load_async_to_lds_b128 %0, %1, off"
                 :: "v"(lds_byte_addr), "v"(gsrc)
                 : "memory");
}
__device__ __forceinline__ void asyncWaitAll() {
    asm volatile("s_wait_asynccnt 0" ::: "memory");
}

// ---------------------------------------------------------------------------
// in-degree count: one thread per edge
// ---------------------------------------------------------------------------
__global__ void count_deg_kernel(const int* __restrict__ dst, float* __restrict__ deg, int E) {
    int e = blockIdx.x * blockDim.x + threadIdx.x;
    if (e < E) atomAddF(&deg[dst[e]], 1.0f);
}

__global__ void invert_deg_kernel(float* __restrict__ deg, int n) {
    int i = blockIdx.x * blockDim.x + threadIdx.x;
    if (i < n) deg[i] = 1.0f / fmaxf(deg[i], 1.0f);
}

// ---------------------------------------------------------------------------
// combine parallel-linear weights: Wc = Wr + Wlin, bc = b + blin (both layers)
// ---------------------------------------------------------------------------
__global__ void combine_kernel(const float* __restrict__ Wr1, const float* __restrict__ Wlin1,
                               const float* __restrict__ b1,  const float* __restrict__ blin1,
                               const float* __restrict__ Wr2, const float* __restrict__ Wlin2,
                               const float* __restrict__ b2,  const float* __restrict__ blin2,
                               float* __restrict__ Wc1, float* __restrict__ Wc2,
                               float* __restrict__ bc1, float* __restrict__ bc2) {
    int i = blockIdx.x * blockDim.x + threadIdx.x;
    if (i < D * D) {
        Wc1[i] = Wr1[i] + Wlin1[i];
        Wc2[i] = Wr2[i] + Wlin2[i];
    }
    if (i < D) {
        bc1[i] = b1[i] + blin1[i];
        bc2[i] = b2[i] + blin2[i];
    }
}

// ---------------------------------------------------------------------------
// scatter-add: acc[dst[e], :] += feat[src[e], :]
// 64 threads per edge, float4 per thread (coalesced 1KB per edge)
// ---------------------------------------------------------------------------
__global__ void scatter_add_kernel(const float* __restrict__ feat,
                                   const int* __restrict__ src,
                                   const int* __restrict__ dst,
                                   float* __restrict__ acc,
                                   long long nwork) {
    long long t = (long long)blockIdx.x * blockDim.x + threadIdx.x;
    if (t >= nwork) return;
    int e = (int)(t >> 6);          // edge id
    int c = (int)(t & 63) << 2;     // feature offset (0..252 step 4)
    int s = src[e];
    int d = dst[e];
    const float4 v = *(const float4*)(feat + (size_t)s * D + c);
    float* p = acc + (size_t)d * D + c;
    atomAddF(p + 0, v.x);
    atomAddF(p + 1, v.y);
    atomAddF(p + 2, v.z);
    atomAddF(p + 3, v.w);
}

// ---------------------------------------------------------------------------
// fused dual-GEMM with mean-divide folded into A loads:
//   out = act( (Aacc * rdeg[row]) @ Wacc + Ax @ Wx + bias )
// Block: 256 threads = 8 wave32; tile = 128 rows x 128 cols (gridDim.y = 2,
// so A operands are streamed only 2x). Weight panels staged in LDS in
// K-chunks of 64 via async-to-LDS (ASYNCcnt). fp32 WMMA 16x16x4.
// ---------------------------------------------------------------------------
__global__ __launch_bounds__(256) void gemm_fused_kernel(
        const float* __restrict__ Aacc, const float* __restrict__ Ax,
        const float* __restrict__ Wacc, const float* __restrict__ Wx,
        const float* __restrict__ bias, const float* __restrict__ rdeg,
        float* __restrict__ out, int nrows, int relu) {
    __shared__ float Bs1[KCHUNK * TILE_COLS];   // 32 KB
    __shared__ float Bs2[KCHUNK * TILE_COLS];   // 32 KB

    const int tid = threadIdx.x;
    const int r0  = blockIdx.x * TILE_ROWS;
    const int c0  = blockIdx.y * TILE_COLS;

    const int wave = tid >> 5;       // 0..7 -> 16-row stripe
    const int lane = tid & 31;
    const int lm   = lane & 15;      // M (A frag) / N (B frag) within tile
    const int lh   = lane >> 4;      // half select -> K base offset

    const int m  = r0 + wave * 16 + lm;              // logical A row for this lane
    const int mc = (m < nrows) ? m : (nrows - 1);    // clamped (EXEC must stay full)
    const float rd = rdeg[mc];
    const float* arow = Aacc + (size_t)mc * D;
    const float* xrow = Ax   + (size_t)mc * D;

    v8f acc[8];
    #pragma unroll
    for (int t = 0; t < 8; ++t) acc[t] = (v8f){0.f, 0.f, 0.f, 0.f, 0.f, 0.f, 0.f, 0.f};

    for (int kc = 0; kc < D / KCHUNK; ++kc) {
        __syncthreads();    // previous chunk fully consumed before overwrite

        // stage 64x128 panels of both weight matrices into LDS (async, 16B/lane)
        for (int i = tid; i < KCHUNK * (TILE_COLS / 4); i += 256) {   // 2048 float4 each
            const int kl = i >> 5;            // local K row   0..63
            const int n4 = i & 31;            // float4 column 0..31
            const int kg = kc * KCHUNK + kl;  // global K row
            const float* g1 = Wacc + (size_t)kg * D + c0 + n4 * 4;
            const float* g2 = Wx   + (size_t)kg * D + c0 + n4 * 4;
            asyncCopy16((unsigned)(uintptr_t)(Bs1 + kl * TILE_COLS + n4 * 4), g1);
            asyncCopy16((unsigned)(uintptr_t)(Bs2 + kl * TILE_COLS + n4 * 4), g2);
        }
        asyncWaitAll();     // this wave's async copies landed in LDS
        __syncthreads();    // all waves' copies visible

        for (int ks = 0; ks < KCHUNK / 4; ++ks) {
            const int klg = kc * KCHUNK + ks * 4 + lh * 2;   // global K (A frag)
            const int kll = ks * 4 + lh * 2;                 // local K  (B frag)

            v2f a1, a2;
            a1.x = arow[klg] * rd;  a1.y = arow[klg + 1] * rd;   // mean-agg, scaled
            a2.x = xrow[klg];       a2.y = xrow[klg + 1];        // root/linear input

            #pragma unroll
            for (int t = 0; t < 8; ++t) {
                v2f b;
                b.x = Bs1[kll * TILE_COLS + t * 16 + lm];
                b.y = Bs1[(kll + 1) * TILE_COLS + t * 16 + lm];
                acc[t] = __builtin_amdgcn_wmma_f32_16x16x4_f32(
                    false, a1, false, b, (short)0, acc[t], false, false);
                b.x = Bs2[kll * TILE_COLS + t * 16 + lm];
                b.y = Bs2[(kll + 1) * TILE_COLS + t * 16 + lm];
                acc[t] = __builtin_amdgcn_wmma_f32_16x16x4_f32(
                    false, a2, false, b, (short)0, acc[t], false, false);
            }
        }
    }

    // epilogue: bias (+relu), store. C/D layout: M = v + 8*lh, N = lm.
    #pragma unroll
    for (int t = 0; t < 8; ++t) {
        const int n  = c0 + t * 16 + lm;
        const float bv = bias[n];
        #pragma unroll
        for (int v = 0; v < 8; ++v) {
            const int row = r0 + wave * 16 + v + lh * 8;
            if (row < nrows) {
                float val = acc[t][v] + bv;
                if (relu) val = fmaxf(val, 0.0f);
                out[(size_t)row * D + n] = val;
            }
        }
    }
}

// ---------------------------------------------------------------------------
// host-side launch
// ---------------------------------------------------------------------------
extern "C" void kernel_launch(void* const* d_in, const int* in_sizes, int n_in,
                              void* d_out, int out_size, void* d_ws, size_t ws_size,
                              hipStream_t stream) {
    const float* x     = (const float*)d_in[0];
    const int*   edges = (const int*)d_in[1];
    const float* Wl1   = (const float*)d_in[2];
    const float* Wr1   = (const float*)d_in[3];
    const float* b1    = (const float*)d_in[4];
    const float* Wlin1 = (const float*)d_in[5];
    const float* blin1 = (const float*)d_in[6];
    const float* Wl2   = (const float*)d_in[7];
    const float* Wr2   = (const float*)d_in[8];
    const float* b2    = (const float*)d_in[9];
    const float* Wlin2 = (const float*)d_in[10];
    const float* blin2 = (const float*)d_in[11];

    const int N = in_sizes[0] / D;
    const int E = in_sizes[1] / 2;
    const int* src = edges;
    const int* dst = edges + E;

    // workspace layout
    float* ACC  = (float*)d_ws;                 // N*D   scatter accumulator (reused)
    float* H    = ACC  + (size_t)N * D;         // N*D   hidden layer output
    float* RD   = H    + (size_t)N * D;         // N     degree -> 1/max(deg,1)
    float* Wc1  = RD   + N;                     // D*D
    float* Wc2  = Wc1  + D * D;                 // D*D
    float* bc1  = Wc2  + D * D;                 // D
    float* bc2  = bc1  + D;                     // D

    const long long scatterWork = (long long)E * (D / 4);
    const int scatterBlocks = (int)((scatterWork + 255) / 256);
    dim3 gemmGrid((N + TILE_ROWS - 1) / TILE_ROWS, D / TILE_COLS);

    // ---- degree (shared by both layers) ----
    hipMemsetAsync(RD, 0, (size_t)N * sizeof(float), stream);
    count_deg_kernel<<<(E + 255) / 256, 256, 0, stream>>>(dst, RD, E);
    invert_deg_kernel<<<(N + 255) / 256, 256, 0, stream>>>(RD, N);

    // ---- combined weights for the parallel Linear branch ----
    combine_kernel<<<(D * D + 255) / 256, 256, 0, stream>>>(
        Wr1, Wlin1, b1, blin1, Wr2, Wlin2, b2, blin2, Wc1, Wc2, bc1, bc2);

    // ---- layer 1: aggregate + fused GEMM (+relu) ----
    hipMemsetAsync(ACC, 0, (size_t)N * D * sizeof(float), stream);
    scatter_add_kernel<<<scatterBlocks, 256, 0, stream>>>(x, src, dst, ACC, scatterWork);
    gemm_fused_kernel<<<gemmGrid, 256, 0, stream>>>(ACC, x, Wl1, Wc1, bc1, RD, H, N, 1);

    // ---- layer 2: aggregate + fused GEMM (no relu) ----
    hipMemsetAsync(ACC, 0, (size_t)N * D * sizeof(float), stream);
    scatter_add_kernel<<<scatterBlocks, 256, 0, stream>>>(H, src, dst, ACC, scatterWork);
    gemm_fused_kernel<<<gemmGrid, 256, 0, stream>>>(ACC, H, Wl2, Wc2, bc2, RD, (float*)d_out, N, 0);
}